// CAMoE_GNN_Layer_7086696038965
// MI455X (gfx1250) — compile-verified
//
#include <hip/hip_runtime.h>
#include <hip/hip_bf16.h>

// ---------------------------------------------------------------------------
// CAMoE GCN layer for MI455X (gfx1250, wave32).
//
// out = sum_i gate[:,i] * relu( (A_hat @ x) @ W_i + b_i )
// using linearity of the GCN aggregation to scatter ONCE on x instead of 3x
// on xW.  Scatter is the memory-bound part (~0.8 GB moved -> ~50us roofline);
// the expert GEMMs run on V_WMMA_F32_16X16X4_F32 (f32 matches reference
// numerics) with W staged in LDS and 4 independent accumulator chains.
// ---------------------------------------------------------------------------

#define DIN   64
#define DOUT  64
#define NEXP  3
#define GATEC 4
#define TEMP_INV (1.0f / 101.0f)

// LDS layout for W: transposed + padded  sW[e][n][k] at e*LDS_E + n*LDS_ROW + k
// row stride 66 so (k,k+1) pairs are one b64 load and half-waves spread banks.
#define LDS_ROW 66
#define LDS_E   (DOUT * LDS_ROW)      // 4224 floats per expert

typedef __attribute__((ext_vector_type(2))) float v2f;
typedef __attribute__((ext_vector_type(8))) float v8f;

__device__ __forceinline__ void atomic_add_f32(float* p, float v) {
    // no-return f32 atomic add (tracked on STOREcnt); avoids CAS fallback
    asm volatile("global_atomic_add_f32 %0, %1, off" :: "v"(p), "v"(v) : "memory");
}

// ---- 1) degree with self loops --------------------------------------------
__global__ __launch_bounds__(256) void deg_init_kernel(int* __restrict__ deg, int n) {
    int i = blockIdx.x * blockDim.x + threadIdx.x;
    if (i < n) deg[i] = 1;   // self loop
}

__global__ __launch_bounds__(256) void deg_count_kernel(const int* __restrict__ cols,
                                                        int* __restrict__ deg, int e) {
    int i = blockIdx.x * blockDim.x + threadIdx.x;
    if (i < e) atomicAdd(&deg[cols[i]], 1);
}

__global__ __launch_bounds__(256) void dinv_kernel(const int* __restrict__ deg,
                                                   float* __restrict__ dinv, int n) {
    int i = blockIdx.x * blockDim.x + threadIdx.x;
    if (i >= n) return;
    float d = (float)deg[i];
    dinv[i] = d > 0.0f ? rsqrtf(d) : 0.0f;
}

// ---- 2) gating softmax -----------------------------------------------------
__global__ __launch_bounds__(256) void gate_kernel(const float* __restrict__ gf,
                                                   const float* __restrict__ Wg,  // [GATEC, NEXP]
                                                   float* __restrict__ gate, int n) {
    int i = blockIdx.x * blockDim.x + threadIdx.x;
    if (i >= n) return;
    float f0 = gf[i * 4 + 0], f1 = gf[i * 4 + 1], f2 = gf[i * 4 + 2], f3 = gf[i * 4 + 3];
    float l[NEXP];
#pragma unroll
    for (int e = 0; e < NEXP; ++e)
        l[e] = (f0 * Wg[0 * NEXP + e] + f1 * Wg[1 * NEXP + e] +
                f2 * Wg[2 * NEXP + e] + f3 * Wg[3 * NEXP + e]) * TEMP_INV;
    float mx = fmaxf(l[0], fmaxf(l[1], l[2]));
    float e0 = __expf(l[0] - mx), e1 = __expf(l[1] - mx), e2 = __expf(l[2] - mx);
    float inv = 1.0f / (e0 + e1 + e2);
    gate[i * 3 + 0] = e0 * inv;
    gate[i * 3 + 1] = e1 * inv;
    gate[i * 3 + 2] = e2 * inv;
}

// ---- 3) h = dinv^2 * x (self loop term; deterministically inits h) ---------
__global__ __launch_bounds__(256) void h_init_kernel(const float* __restrict__ x,
                                                     const float* __restrict__ dinv,
                                                     float* __restrict__ h, int n) {
    int t = blockIdx.x * blockDim.x + threadIdx.x;   // n * 16 threads (float4 each)
    int node = t >> 4;
    if (node >= n) return;
    int c4 = (t & 15) * 4;
    float w = dinv[node];
    w = w * w;
    float4 v = *(const float4*)(x + (size_t)node * DIN + c4);
    float4 o;
    o.x = v.x * w; o.y = v.y * w; o.z = v.z * w; o.w = v.w * w;
    *(float4*)(h + (size_t)node * DIN + c4) = o;
}

// ---- 4) edge scatter: h[col] += dinv[r]*dinv[c] * x[row] -------------------
__global__ __launch_bounds__(256) void scatter_kernel(const float* __restrict__ x,
                                                      const int* __restrict__ rows,
                                                      const int* __restrict__ cols,
                                                      const float* __restrict__ dinv,
                                                      float* __restrict__ h, int e) {
    int t = blockIdx.x * blockDim.x + threadIdx.x;   // e * 16 threads
    int edge = t >> 4;
    if (edge >= e) return;
    int c4 = (t & 15) * 4;
    int r = rows[edge], c = cols[edge];
    float w = dinv[r] * dinv[c];
    float4 xv = *(const float4*)(x + (size_t)r * DIN + c4);
    float* hp = h + (size_t)c * DIN + c4;
    atomic_add_f32(hp + 0, w * xv.x);
    atomic_add_f32(hp + 1, w * xv.y);
    atomic_add_f32(hp + 2, w * xv.z);
    atomic_add_f32(hp + 3, w * xv.w);
}

// ---- 5) fused MoE GEMM epilogue on f32 WMMA --------------------------------
// One wave owns a 16-row tile of h.  W (48KB) is staged transposed in LDS by
// the whole block; each B-fragment is then one ds_load_b64.  Loop order is
// kt-outer / ntile-inner with 4 independent accumulator chains so LDS loads
// for step kt+1 overlap the 4 WMMAs of step kt.
//
// f32 WMMA operand layout (ISA 7.12.2):
//   A 16x4:  lane m=l&15; VGPR0=K(2*(l>>4)), VGPR1=K(2*(l>>4)+1)
//   B 4x16:  lane n=l&15; same K striping
//   C 16x16: lane col n=l&15; VGPR v holds row  v + 8*(l>>4)
__global__ __launch_bounds__(256) void moe_wmma_kernel(const float* __restrict__ h,    // [N,64]
                                                       const float* __restrict__ W,    // [3,64,64]
                                                       const float* __restrict__ bias, // [3,64]
                                                       const float* __restrict__ gate, // [N,3]
                                                       float* __restrict__ out,        // [N,64]
                                                       int nNodes) {
    __shared__ float sW[NEXP * LDS_E];   // ~50.7 KB, transposed W
    __shared__ float sB[NEXP * DOUT];

    // cooperative stage: sW[e][n][k] = W[e][k][n]   (coalesced global reads)
    for (int idx = threadIdx.x; idx < NEXP * DIN * DOUT; idx += 256) {
        int e   = idx >> 12;         // / 4096
        int rem = idx & 4095;
        int k   = rem >> 6;
        int ncl = rem & 63;
        sW[e * LDS_E + ncl * LDS_ROW + k] = W[idx];
    }
    for (int idx = threadIdx.x; idx < NEXP * DOUT; idx += 256) sB[idx] = bias[idx];
    __syncthreads();

    const int lane = threadIdx.x & 31;
    const int wave = threadIdx.x >> 5;
    const int rowBase = (blockIdx.x * 8 + wave) * 16;
    if (rowBase >= nNodes) return;       // wave-uniform: EXEC stays all-ones for WMMA
    const int m  = lane & 15;            // A row / B,C column owned by this lane
    const int hi = lane >> 4;            // half-wave select
    const int kb = hi * 2;               // K sub-offset within a K=4 step

    // A fragments of this wave's 16x64 h tile: 16 steps of K=4  (32 VGPRs)
    const float* hrow = h + (size_t)(rowBase + m) * DIN;
    v2f afr[16];
#pragma unroll
    for (int kt = 0; kt < 16; ++kt) {
        float2 t2 = *(const float2*)(hrow + kt * 4 + kb);
        afr[kt].x = t2.x;
        afr[kt].y = t2.y;
    }

    v8f acc[4] = {};                     // 16x64 output tile accumulator

#pragma unroll
    for (int e = 0; e < NEXP; ++e) {
        // gate values for the 8 rows held in this lane's C registers
        float g[8];
#pragma unroll
        for (int v = 0; v < 8; ++v)
            g[v] = gate[(size_t)(rowBase + v + 8 * hi) * 3 + e];

        const float* sWe = sW + e * LDS_E;
        v8f c[4] = {};
#pragma unroll
        for (int kt = 0; kt < 16; ++kt) {
            const int k = kt * 4 + kb;
#pragma unroll
            for (int t = 0; t < 4; ++t) {
                const int ncol = t * 16 + m;
                float2 t2 = *(const float2*)(sWe + ncol * LDS_ROW + k);  // ds_load_b64
                v2f bfr;
                bfr.x = t2.x;
                bfr.y = t2.y;
                c[t] = __builtin_amdgcn_wmma_f32_16x16x4_f32(
                           /*neg_a=*/false, afr[kt], /*neg_b=*/false, bfr,
                           /*c_mod=*/(short)0, c[t], /*reuse_a=*/false, /*reuse_b=*/false);
            }
        }
        // epilogue: relu(C + bias) * gate, accumulated over experts
#pragma unroll
        for (int t = 0; t < 4; ++t) {
            const float bv = sB[e * DOUT + t * 16 + m];
#pragma unroll
            for (int v = 0; v < 8; ++v) {
                float z = c[t][v] + bv;
                z = z > 0.0f ? z : 0.0f;
                acc[t][v] += g[v] * z;
            }
        }
    }

    // write out (N is a multiple of 16, so tiles are always full)
#pragma unroll
    for (int t = 0; t < 4; ++t)
#pragma unroll
        for (int v = 0; v < 8; ++v)
            out[(size_t)(rowBase + v + 8 * hi) * DOUT + t * 16 + m] = acc[t][v];
}

// ---------------------------------------------------------------------------
extern "C" void kernel_launch(void* const* d_in, const int* in_sizes, int n_in,
                              void* d_out, int out_size, void* d_ws, size_t ws_size,
                              hipStream_t stream) {
    const float* x    = (const float*)d_in[0];   // [N,64]
    const int*   ei   = (const int*)  d_in[1];   // [2,E]
    const float* gf   = (const float*)d_in[2];   // [N,4]
    const float* W    = (const float*)d_in[3];   // [3,64,64]
    const float* bias = (const float*)d_in[4];   // [3,64]
    const float* Wg   = (const float*)d_in[5];   // [4,3]

    const int N = in_sizes[0] / DIN;
    const int E = in_sizes[1] / 2;
    const int* rows = ei;
    const int* cols = ei + E;

    // workspace layout (~27.6 MB)
    char*  ws   = (char*)d_ws;
    float* h    = (float*)ws;                       size_t off = (size_t)N * DIN * sizeof(float);
    int*   deg  = (int*)(ws + off);                 off += (size_t)N * sizeof(int);
    float* dinv = (float*)(ws + off);               off += (size_t)N * sizeof(float);
    float* gate = (float*)(ws + off);

    float* out = (float*)d_out;

    const int B = 256;
    deg_init_kernel <<<(N + B - 1) / B, B, 0, stream>>>(deg, N);
    deg_count_kernel<<<(E + B - 1) / B, B, 0, stream>>>(cols, deg, E);
    dinv_kernel     <<<(N + B - 1) / B, B, 0, stream>>>(deg, dinv, N);
    gate_kernel     <<<(N + B - 1) / B, B, 0, stream>>>(gf, Wg, gate, N);
    h_init_kernel   <<<(N * 16 + B - 1) / B, B, 0, stream>>>(x, dinv, h, N);
    scatter_kernel  <<<(E * 16 + B - 1) / B, B, 0, stream>>>(x, rows, cols, dinv, h, E);
    moe_wmma_kernel <<<(N + 127) / 128, B, 0, stream>>>(h, W, bias, gate, out, N);
}